// FwdModel_27779848470584
// MI455X (gfx1250) — compile-verified
//
#include <hip/hip_runtime.h>
#include <stdint.h>

// ---------------------------------------------------------------------------
// Depth-dependent PSF blur, restructured as ONE spatially-varying 71x71
// stencil per output pixel:
//   interior tap -> weight = psf[ch,ky,kx, psi(y,x)]
//   border  tap -> weight = sum over present classes of psf[ch,ky,kx,c]
//   out = clip(sum(x*w) / sum(w), 0, 1)   (den==0 -> 1)
// 4.2G taps total. LDS-gather bound; WMMA not applicable (per-pixel
// data-dependent weights -> no shared B operand). CDNA5 path used: Tensor
// Data Mover (tensor_load_to_lds, TENSORcnt) for async LDS staging.
// ---------------------------------------------------------------------------

#define KSZ   71
#define PADR  35
#define NPSF  3
#define NCLS  15
#define NB    8
#define HH    256
#define WW    256
#define SHH   186
#define SWW   186
#define TILE  16
#define HALO  (KSZ - 1)        // 70
#define TDIM  (TILE + HALO)    // 86
#define LROW  (16 * KSZ)       // 1136 floats per ky row, layout [c][kx]
#define KC    2                // ky rows per LUT chunk (double buffered)
#define NTHR  256

typedef unsigned int       u32;
typedef unsigned long long u64;
typedef unsigned char      u8;

// -------- TDM support (this toolchain: 6-arg builtin, therock-10.0) --------
#if defined(__has_builtin)
#  if __has_builtin(__builtin_amdgcn_tensor_load_to_lds) && \
      __has_builtin(__builtin_amdgcn_s_wait_tensorcnt)
#    define HAVE_TDM 1
#  endif
#endif
#ifndef HAVE_TDM
#  define HAVE_TDM 0
#endif

#if HAVE_TDM
typedef __attribute__((ext_vector_type(4))) u32 u32x4_t;
typedef __attribute__((ext_vector_type(8))) int i32x8_t;
typedef __attribute__((ext_vector_type(4))) int i32x4_t;

// Build D# per CDNA5 ISA 8.3/8.4 and issue TENSOR_LOAD_TO_LDS.
// 2D tile: tile_d0 elements per row, tile_d1 rows, row stride in elements.
__device__ __forceinline__ void tdm_issue(u32 lds_off, u64 gaddr, u32 dsz_code,
                                          u32 tile_d0, u32 tile_d1,
                                          u64 row_stride_elems) {
  u32x4_t g0;
  g0.x = 0x1u;                                        // count=1, user mode
  g0.y = lds_off;                                     // lds_addr [63:32]
  g0.z = (u32)(gaddr & 0xFFFFFFFFull);                // global_addr lo
  g0.w = (u32)((gaddr >> 32) & 0x01FFFFFFull) | (2u << 30); // hi + type=2
  const u32 td0 = 1u << 20, td1 = 1u << 20;           // tensor dims: never OOB
  i32x8_t g1;
  g1[0] = (int)(dsz_code << 16);                      // data_size, mask=0
  g1[1] = (int)((td0 & 0xFFFFu) << 16);               // tensor_dim0[15:0]
  g1[2] = (int)(((td0 >> 16) & 0xFFFFu) | ((td1 & 0xFFFFu) << 16));
  g1[3] = (int)(((td1 >> 16) & 0xFFFFu) | ((tile_d0 & 0xFFFFu) << 16));
  g1[4] = (int)(tile_d1 & 0xFFFFu);                   // tile_dim1, tile_dim2=0
  g1[5] = (int)(u32)(row_stride_elems & 0xFFFFFFFFull);
  g1[6] = (int)(u32)((row_stride_elems >> 32) & 0xFFFFull);
  g1[7] = 0;
  i32x4_t gz4 = {0, 0, 0, 0};                         // groups 2/3 unused (2D)
  i32x8_t gz8 = {0, 0, 0, 0, 0, 0, 0, 0};             // extra arg (6-arg form)
  __builtin_amdgcn_tensor_load_to_lds(g0, g1, gz4, gz4, gz8, 0);
}
#endif

// ---------------------- Phase 1a: per-sample presence -----------------------
__global__ void build_present(const int* __restrict__ psi, u32* __restrict__ pres) {
  __shared__ u32 m;
  if (threadIdx.x == 0) m = 0u;
  __syncthreads();
  const int n = blockIdx.x;
  u32 local = 0u;
  for (int i = threadIdx.x; i < SHH * SWW; i += blockDim.x)
    local |= (1u << (psi[(size_t)n * SHH * SWW + i] & 31));
  atomicOr(&m, local);
  __syncthreads();
  if (threadIdx.x == 0) pres[n] = m & 0x7FFFu;
}

// ------------- Phase 1b: padded class map CLS[n,256,256] (border=15) --------
__global__ void build_cls(const int* __restrict__ psi, u8* __restrict__ cls) {
  int idx = blockIdx.x * blockDim.x + threadIdx.x;
  if (idx >= NB * HH * WW) return;
  int n = idx >> 16;
  int rem = idx & 0xFFFF;
  int y = rem >> 8, x = rem & 255;
  u8 v = 15;
  if (y >= PADR && y < PADR + SHH && x >= PADR && x < PADR + SWW)
    v = (u8)(psi[((size_t)n * SHH + (y - PADR)) * SWW + (x - PADR)] & 15);
  cls[idx] = v;
}

// --- Phase 1c: LUT[n,ch,ky,c,kx] f32; slot c==15 = per-sample present-sum ---
__global__ void build_lut(const float* __restrict__ psf, const u32* __restrict__ pres,
                          float* __restrict__ lut) {
  int idx = blockIdx.x * blockDim.x + threadIdx.x;  // one thread per (n,ch,ky,c)
  const int TOT = NB * NPSF * KSZ * 16;
  if (idx >= TOT) return;
  int cc = idx & 15;
  int t  = idx >> 4;
  int ky = t % KSZ;
  int nc = t / KSZ;                // n*3 + ch
  int ch = nc % NPSF;
  int n  = nc / NPSF;
  const float* prow = psf + ((size_t)(ch * KSZ + ky) * KSZ) * NCLS; // +kx*15+c
  float* dst = lut + ((size_t)((size_t)nc * KSZ + ky) * 16 + cc) * KSZ;
  if (cc < NCLS) {
    for (int kx = 0; kx < KSZ; ++kx) dst[kx] = prow[kx * NCLS + cc];
  } else {
    u32 m = pres[n];
    for (int kx = 0; kx < KSZ; ++kx) {
      float s = 0.f;
      const float* pp = prow + kx * NCLS;
      for (int q = 0; q < NCLS; ++q)
        if ((m >> q) & 1u) s += pp[q];
      dst[kx] = s;
    }
  }
}

// ------------------------------- Main kernel --------------------------------
__global__ __launch_bounds__(NTHR)
void psf_blur_kernel(const float* __restrict__ x,
                     const u8*   __restrict__ cls,
                     const float* __restrict__ lut,
                     float* __restrict__ out) {
  __shared__ float xT[TDIM * TDIM];      // 29584 B
  __shared__ u8    clsT[TDIM * TDIM];    //  7396 B
  __shared__ float lutB[2][KC * LROW];   // 18176 B  (total ~55 KB)

  const int tid  = threadIdx.x;
  const int tile = blockIdx.x;
  const int ch   = blockIdx.y;
  const int n    = blockIdx.z;
  const int oy0  = min((tile / 12) * TILE, SHH - TILE);
  const int ox0  = min((tile % 12) * TILE, SWW - TILE);

  const float* xplane = x   + (size_t)(n * NPSF + ch) * HH * WW;
  const u8*    cplane = cls + (size_t)n * HH * WW;
  const float* lslab  = lut + (size_t)(n * NPSF + ch) * KSZ * LROW;

  const int NCHUNK = (KSZ + KC - 1) / KC;   // 36

  // ---- prologue: stage x tile, class tile, LUT chunk 0 ----
#if HAVE_TDM
  if (tid < 32) {
    tdm_issue((u32)(uintptr_t)xT,
              (u64)(uintptr_t)(xplane + (size_t)oy0 * WW + ox0), 2u, TDIM, TDIM, WW);
    tdm_issue((u32)(uintptr_t)clsT,
              (u64)(uintptr_t)(cplane + (size_t)oy0 * WW + ox0), 0u, TDIM, TDIM, WW);
    tdm_issue((u32)(uintptr_t)&lutB[0][0],
              (u64)(uintptr_t)lslab, 2u, KC * LROW, 1u, KC * LROW);
  }
#else
  for (int i = tid; i < TDIM * TDIM; i += NTHR) {
    int yy = i / TDIM, xx = i - yy * TDIM;
    xT[i]   = xplane[(size_t)(oy0 + yy) * WW + ox0 + xx];
    clsT[i] = cplane[(size_t)(oy0 + yy) * WW + ox0 + xx];
  }
  for (int i = tid; i < KC * LROW; i += NTHR) lutB[0][i] = lslab[i];
#endif

  const int ty = tid >> 4, tx = tid & 15;
  float num = 0.f, den = 0.f;

  for (int c = 0; c < NCHUNK; ++c) {
#if HAVE_TDM
    if (tid < 32) __builtin_amdgcn_s_wait_tensorcnt(0);  // chunk c (and tiles) ready
#endif
    __syncthreads();   // all waves: buf[c&1] valid; buf[(c+1)&1] free to refill

    const int r0  = c * KC;
    const int rn  = min(KC, KSZ - r0);
    const int buf = c & 1;

    if (c + 1 < NCHUNK) {   // prefetch next LUT chunk into the other buffer
      const int r0n = r0 + KC;
      const int cnt = min(KC, KSZ - r0n) * LROW;
#if HAVE_TDM
      if (tid < 32)
        tdm_issue((u32)(uintptr_t)&lutB[buf ^ 1][0],
                  (u64)(uintptr_t)(lslab + (size_t)r0n * LROW), 2u, cnt, 1u, cnt);
#else
      const float* src = lslab + (size_t)r0n * LROW;
      float* dst = &lutB[buf ^ 1][0];
      for (int i = tid; i < cnt; i += NTHR) dst[i] = src[i];
#endif
    }

    // ---- consume current chunk: rn ky-rows x 71 kx taps ----
    for (int r = 0; r < rn; ++r) {
      const int ky = r0 + r;
      const float* lrow = &lutB[buf][r * LROW];   // [c][kx], c-stride 71 -> no bank conflicts
      const int pb = (ty + ky) * TDIM + tx;
      #pragma unroll 4
      for (int kx = 0; kx < KSZ; ++kx) {
        const int p = pb + kx;
        const u32 cc = clsT[p];                   // ds_load_u8
        const float w = lrow[cc * KSZ + kx];      // ds_load_b32 gather
        num = fmaf(xT[p], w, num);                // ds_load_b32 + v_fmac
        den += w;
      }
    }
  }

  if (den == 0.f) den = 1.f;
  float r = num / den;
  r = fminf(fmaxf(r, 0.f), 1.f);
  out[(((size_t)n * NPSF + ch) * SHH + (oy0 + ty)) * SWW + (ox0 + tx)] = r;
}

// ------------------------------- launcher -----------------------------------
extern "C" void kernel_launch(void* const* d_in, const int* in_sizes, int n_in,
                              void* d_out, int out_size, void* d_ws, size_t ws_size,
                              hipStream_t stream) {
  (void)in_sizes; (void)n_in; (void)out_size; (void)ws_size;
  const float* x   = (const float*)d_in[0];   // [8,3,256,256] f32
  const int*   psi = (const int*)d_in[1];     // [8,186,186]   i32
  const float* psf = (const float*)d_in[2];   // [3,71,71,15]  f32
  float* out = (float*)d_out;                 // [8,3,186,186] f32

  char* ws = (char*)d_ws;                     // layout (8.27 MB total):
  u32*  pres = (u32*)ws;                      //   [0,32)      presence masks
  u8*   cls  = (u8*)(ws + 256);               //   [256, +512K) padded class map
  float* lut = (float*)(ws + 256 + (size_t)NB * HH * WW);  // 7.74 MB LUT

  build_present<<<NB, 256, 0, stream>>>(psi, pres);
  build_cls<<<(NB * HH * WW) / 256, 256, 0, stream>>>(psi, cls);
  build_lut<<<(NB * NPSF * KSZ * 16 + 255) / 256, 256, 0, stream>>>(psf, pres, lut);
  psf_blur_kernel<<<dim3(12 * 12, NPSF, NB), NTHR, 0, stream>>>(x, cls, lut, out);
}